// Transformer_2D_14602888806566
// MI455X (gfx1250) — compile-verified
//
#include <hip/hip_runtime.h>
#include <stdint.h>

// Problem constants (from the reference): src [8,16,512,512] f32, flow [8,2,512,512] f32
#define B_  8
#define C_  16
#define H_  512
#define W_  512
#define HW_ (H_ * W_)          // 262144 = 2^18

// Bilinear flow-warp (grid_sample, align_corners=True, padding=border).
// One thread = one (b,h,w) pixel, all 16 channels.
// CDNA5 specifics:
//  - flow values staged LDS-side via global_load_async_to_lds_b32 (ASYNCcnt)
//  - output written with non-temporal stores so the 128MB src stays resident
//    in the 192MB L2 while the 128MB output streams through.
__global__ __launch_bounds__(256) void warp_bilinear_kernel(
    const float* __restrict__ src,
    const float* __restrict__ flow,
    float* __restrict__ out)
{
    __shared__ float sflow[2 * 256];   // [0..255] = flow_y, [256..511] = flow_x

    const int tid = threadIdx.x;
    const int p   = blockIdx.x * 256 + tid;   // global pixel id in [0, B*H*W)
    const int b   = p >> 18;                  // / (H*W)
    const int pin = p & (HW_ - 1);            // pixel index within batch image
    const int h   = pin >> 9;                 // / W
    const int w   = pin & (W_ - 1);

    // ---- async-stage the 2 flow values for this lane into LDS (gfx1250 path)
    {
        const float* f0 = flow + (size_t)b * 2 * HW_ + pin;  // flow[b,0,h,w]
        const float* f1 = f0 + HW_;                          // flow[b,1,h,w]
        // Low 32 bits of a generic LDS pointer are the LDS byte offset.
        uint32_t lds0 = (uint32_t)(uintptr_t)(&sflow[tid]);
        uint32_t lds1 = (uint32_t)(uintptr_t)(&sflow[256 + tid]);
        uint64_t a0 = (uint64_t)(uintptr_t)f0;
        uint64_t a1 = (uint64_t)(uintptr_t)f1;
        asm volatile(
            "global_load_async_to_lds_b32 %0, %2, off\n\t"
            "global_load_async_to_lds_b32 %1, %3, off\n\t"
            "s_wait_asynccnt 0x0"
            :
            : "v"(lds0), "v"(lds1), "v"(a0), "v"(a1)
            : "memory");
    }
    const float fy = sflow[tid];        // displacement along H (row)
    const float fx = sflow[256 + tid];  // displacement along W (col)

    // ---- coordinates, mirroring the reference math exactly
    const float new_y = (float)h + fy;
    const float new_x = (float)w + fx;
    const float ny = 2.0f * (new_y / (float)(H_ - 1) - 0.5f);
    const float nx = 2.0f * (new_x / (float)(W_ - 1) - 0.5f);
    float py = (ny + 1.0f) * 0.5f * (float)(H_ - 1);
    float px = (nx + 1.0f) * 0.5f * (float)(W_ - 1);
    py = fminf(fmaxf(py, 0.0f), (float)(H_ - 1));   // padding_mode = border
    px = fminf(fmaxf(px, 0.0f), (float)(W_ - 1));

    const float y0f = floorf(py);
    const float x0f = floorf(px);
    const float wy = py - y0f;
    const float wx = px - x0f;

    int y0 = (int)y0f;  y0 = min(max(y0, 0), H_ - 1);
    int x0 = (int)x0f;  x0 = min(max(x0, 0), W_ - 1);
    const int y1 = min(y0 + 1, H_ - 1);
    const int x1 = min(x0 + 1, W_ - 1);

    const float w00 = (1.0f - wy) * (1.0f - wx);
    const float w01 = (1.0f - wy) * wx;
    const float w10 = wy * (1.0f - wx);
    const float w11 = wy * wx;

    const int o00 = y0 * W_ + x0;
    const int o01 = y0 * W_ + x1;
    const int o10 = y1 * W_ + x0;
    const int o11 = y1 * W_ + x1;

    const float* sb = src + (size_t)b * C_ * HW_;
    float*       ob = out + (size_t)b * C_ * HW_ + pin;

    // 16 channels, fully unrolled: 64 gather loads in flight, then 16 NT stores.
#pragma unroll
    for (int c = 0; c < C_; ++c) {
        const float* sc = sb + (size_t)c * HW_;
        const float v00 = sc[o00];
        const float v01 = sc[o01];
        const float v10 = sc[o10];
        const float v11 = sc[o11];
        const float r = v00 * w00 + v01 * w01 + v10 * w10 + v11 * w11;
        // Non-temporal: output is write-once/stream; keep src resident in L2.
        __builtin_nontemporal_store(r, ob + (size_t)c * HW_);
    }
}

extern "C" void kernel_launch(void* const* d_in, const int* in_sizes, int n_in,
                              void* d_out, int out_size, void* d_ws, size_t ws_size,
                              hipStream_t stream) {
    (void)in_sizes; (void)n_in; (void)out_size; (void)d_ws; (void)ws_size;
    const float* src  = (const float*)d_in[0];
    const float* flow = (const float*)d_in[1];
    float*       out  = (float*)d_out;

    const int total_pixels = B_ * HW_;          // 2,097,152
    const int block = 256;                      // 8 wave32 waves
    const int grid  = total_pixels / block;     // 8192 blocks, exact cover
    warp_bilinear_kernel<<<grid, block, 0, stream>>>(src, flow, out);
}